// SegContrastiveLoss_32933809226342
// MI455X (gfx1250) — compile-verified
//
#include <hip/hip_runtime.h>
#include <hip/hip_bf16.h>

typedef __attribute__((ext_vector_type(2))) float v2f;
typedef __attribute__((ext_vector_type(8))) float v8f;

#define SEG_MARGIN 1.0f

__device__ __forceinline__ float wred_f(float v) {
#pragma unroll
    for (int off = 16; off > 0; off >>= 1) v += __shfl_down(v, off, 32);
    return v;
}
__device__ __forceinline__ unsigned wred_u(unsigned v) {
#pragma unroll
    for (int off = 16; off > 0; off >>= 1) v += __shfl_down(v, off, 32);
    return v;
}

// ---------------------------------------------------------------------------
// Kernel 1: gather sampled features into a compact [B][Sp][C] buffer,
// compute per-row squared norms, copy labels (pad rows -> label 0).
// One wave (32 lanes) per (b, s); each lane handles channels t and t+32.
// Block (0,0) also zeroes the per-batch accumulators.
// ---------------------------------------------------------------------------
__global__ void seg_gather(const float* __restrict__ inputs,
                           const int*   __restrict__ labels,
                           const int*   __restrict__ sidx,
                           float* __restrict__ feats,   // [B][Sp][C]
                           float* __restrict__ n2,      // [B][Sp]
                           int*   __restrict__ labs,    // [B][Sp]
                           float* __restrict__ lossAcc, // [B]
                           unsigned* __restrict__ pairsAcc, // [B]
                           int S, int Sp, int C, long long HW, int B) {
    const int s = blockIdx.x;
    const int b = blockIdx.y;
    const int t = threadIdx.x;

    if (s == 0 && b == 0) {
        if (t < B)              lossAcc[t] = 0.0f;
        else if (t < 2 * B)     pairsAcc[t - B] = 0u;
    }

    float* frow = feats + ((size_t)b * Sp + s) * C;
    float v0 = 0.0f, v1 = 0.0f;
    int lab = 0;
    if (s < S) {
        const long long idx = (long long)sidx[s];
        const float* base = inputs + (long long)b * C * HW + idx;
        v0 = base[(long long)t * HW];
        v1 = base[(long long)(t + 32) * HW];
        lab = labels[(long long)b * HW + idx];
    }
    frow[t]      = v0;
    frow[t + 32] = v1;

    float sq = wred_f(v0 * v0 + v1 * v1);
    if (t == 0) {
        n2[(size_t)b * Sp + s]   = sq;
        labs[(size_t)b * Sp + s] = lab;
    }
}

// ---------------------------------------------------------------------------
// Kernel 2: one wave per 16x16 Gram tile. 16x chained V_WMMA_F32_16X16X4_F32
// accumulate G = F_row x F_col^T over C=64. Epilogue turns G into pairwise
// distances and accumulates the contrastive loss terms + pair counts.
//
// f32 16x4 A fragment lane layout (ISA 7.12.2): lanes 0-15 hold K=0 (VGPR0)
// and K=1 (VGPR1); lanes 16-31 hold K=2,3. B (4x16) mirrors this with the
// lane index as N. C/D: VGPR v, lanes 0-15 -> M=v, N=lane; lanes 16-31 ->
// M=v+8, N=lane-16.
// ---------------------------------------------------------------------------
__global__ void seg_tiles(const float* __restrict__ feats,
                          const float* __restrict__ n2,
                          const int*   __restrict__ labs,
                          float* __restrict__ lossAcc,
                          unsigned* __restrict__ pairsAcc,
                          int Sp, int C, int tilesPerRow) {
    const int b    = blockIdx.y;
    const int tr   = blockIdx.x / tilesPerRow;
    const int tc   = blockIdx.x % tilesPerRow;
    const int lane = threadIdx.x;
    const int half = lane >> 4;   // 0: K pairs {0,1}; 1: K pairs {2,3}
    const int m    = lane & 15;
    const int rowBase = tr * 16;
    const int colBase = tc * 16;

    const float* arow = feats + ((size_t)b * Sp + rowBase + m) * C + half * 2;
    const float* brow = feats + ((size_t)b * Sp + colBase + m) * C + half * 2;

    v8f acc = {0.f, 0.f, 0.f, 0.f, 0.f, 0.f, 0.f, 0.f};
#pragma unroll
    for (int k = 0; k < 16; ++k) {
        v2f a  = *(const v2f*)(arow + 4 * k);
        v2f bb = *(const v2f*)(brow + 4 * k);
        // 8 args: (neg_a, A, neg_b, B, c_mod, C, reuse_a, reuse_b)
        acc = __builtin_amdgcn_wmma_f32_16x16x4_f32(
            false, a, false, bb, (short)0, acc, false, false);
    }

    // Epilogue: distances + margin loss.
    const int j    = colBase + (lane & 15);
    const int labJ = labs[(size_t)b * Sp + j];
    const float nJ = n2[(size_t)b * Sp + j];
    const int rowOff = rowBase + ((lane >= 16) ? 8 : 0);

    float lsum = 0.0f;
    unsigned pcnt = 0u;
#pragma unroll
    for (int v = 0; v < 8; ++v) {
        const int i    = rowOff + v;
        const int labI = labs[(size_t)b * Sp + i];
        const float nI = n2[(size_t)b * Sp + i];
        float d2 = fmaxf(nI + nJ - 2.0f * acc[v], 0.0f);
        float dist = (d2 > 0.0f) ? sqrtf(d2) : 0.0f;
        const bool pv    = (labI != 0) && (labJ != 0);
        const bool same  = (labI == labJ);
        const bool selfm = (dist <= 1e-8f);   // torch.isclose(dist, 0)
        if (pv && same && !selfm) { lsum += 0.5f * dist * dist; pcnt++; }
        if (pv && !same) {
            float tt = fmaxf(SEG_MARGIN - dist, 0.0f);
            lsum += 0.5f * tt * tt;
            pcnt++;
        }
    }

    lsum = wred_f(lsum);
    pcnt = wred_u(pcnt);
    if (lane == 0) {
        atomicAdd(&lossAcc[b], lsum);
        atomicAdd(&pairsAcc[b], pcnt);
    }
}

// ---------------------------------------------------------------------------
// Kernel 3: per-batch gating + final mean (single wave, deterministic).
// ---------------------------------------------------------------------------
__global__ void seg_final(const int* __restrict__ labs,
                          const float* __restrict__ lossAcc,
                          const unsigned* __restrict__ pairsAcc,
                          float* __restrict__ out,
                          int B, int S, int Sp) {
    const int lane = threadIdx.x;
    float totalLoss = 0.0f;
    int vc = 0;
    for (int b = 0; b < B; ++b) {
        unsigned cnt = 0;
        for (int s = lane; s < S; s += 32)
            cnt += (labs[(size_t)b * Sp + s] != 0) ? 1u : 0u;
        cnt = wred_u(cnt);
        if (lane == 0) {
            const unsigned pairs = pairsAcc[b];
            const bool valid = (cnt >= 2u) && (pairs > 0u);
            const unsigned pden = (pairs > 1u) ? pairs : 1u;
            totalLoss += valid ? (lossAcc[b] / (float)pden) : 0.0f;
            vc += valid ? 1 : 0;
        }
    }
    if (lane == 0) {
        const int vden = (vc > 1) ? vc : 1;
        out[0] = (vc > 0) ? (totalLoss / (float)vden) : 0.0f;
    }
}

extern "C" void kernel_launch(void* const* d_in, const int* in_sizes, int n_in,
                              void* d_out, int out_size, void* d_ws, size_t ws_size,
                              hipStream_t stream) {
    const float* inputs = (const float*)d_in[0];   // [B,C,H,W] f32
    const int*   labels = (const int*)d_in[1];     // [B,1,H,W] i32
    const int*   sidx   = (const int*)d_in[2];     // [S] i32
    float* out = (float*)d_out;

    const int B = 4, C = 64, H = 512, W = 512;
    const long long HW = (long long)H * W;
    const int S  = in_sizes[2];                    // 262
    const int Sp = ((S + 15) / 16) * 16;           // 272
    const int tiles = Sp / 16;                     // 17

    // Workspace carve-up (16B aligned chunks).
    char* ws = (char*)d_ws;
    size_t off = 0;
    auto carve = [&](size_t bytes) {
        void* p = ws + off;
        off += (bytes + 15) & ~(size_t)15;
        return p;
    };
    float*    feats    = (float*)carve((size_t)B * Sp * C * sizeof(float)); // 272 KB
    float*    n2       = (float*)carve((size_t)B * Sp * sizeof(float));
    int*      labs     = (int*)  carve((size_t)B * Sp * sizeof(int));
    float*    lossAcc  = (float*)carve((size_t)B * sizeof(float));
    unsigned* pairsAcc = (unsigned*)carve((size_t)B * sizeof(unsigned));
    (void)ws_size;

    seg_gather<<<dim3(Sp, B), 32, 0, stream>>>(inputs, labels, sidx,
                                               feats, n2, labs,
                                               lossAcc, pairsAcc,
                                               S, Sp, C, HW, B);
    seg_tiles<<<dim3(tiles * tiles, B), 32, 0, stream>>>(feats, n2, labs,
                                                         lossAcc, pairsAcc,
                                                         Sp, C, tiles);
    seg_final<<<1, 32, 0, stream>>>(labs, lossAcc, pairsAcc, out, B, S, Sp);
}